// DATT_Module_66546223284567
// MI455X (gfx1250) — compile-verified
//
#include <hip/hip_runtime.h>

typedef float v4f __attribute__((ext_vector_type(4)));

// Problem constants (fixed by the reference: B=8, C=64, T=32, H=W=56, WIN=7)
namespace {
constexpr int Bn  = 8;
constexpr int Cn  = 64;
constexpr int Tn  = 32;
constexpr int HWn = 56 * 56;             // 3136 (divisible by 4)
constexpr int BHWn = Bn * HWn;           // 25088
constexpr int BLK  = 128;                // threads per block (4 waves)
constexpr int VEC  = 4;                  // floats per thread per t
constexpr int POS  = BLK * VEC;          // 512 spatial positions per block
constexpr int GRIDX = BHWn / POS;        // 49, exact
constexpr float kInvCount = 1.0f / (float)(Bn * Tn * HWn); // 1/802816
constexpr float kEps = 1e-5f;
}

// ---- CDNA5 async global->LDS copy (ASYNCcnt-tracked), GV mode, inline asm ----
// Flat LDS pointers carry the LDS byte offset in the low 32 bits (ISA aperture
// rule: LDS address is the low 32 bits); the 64-bit flat pointer is the global
// address for the v[addr] pair. 'off' = no SADDR (GV addressing).
__device__ __forceinline__ void async_copy_b128(const float* gp, float* lp) {
  unsigned lds_off = (unsigned)(unsigned long long)lp;
  unsigned long long gaddr = (unsigned long long)gp;
  asm volatile("global_load_async_to_lds_b128 %0, %1, off"
               :: "v"(lds_off), "v"(gaddr)
               : "memory");
}

template <int N>
__device__ __forceinline__ void wait_asynccnt() {
  asm volatile("s_wait_asynccnt %0" :: "i"(N) : "memory");
}

__device__ __forceinline__ float wave_sum(float v) {
#pragma unroll
  for (int off = 16; off > 0; off >>= 1) v += __shfl_down(v, off, 32);
  return v;
}

// Antisymmetric 7-tap FIR (coeffs [-6,-4,-2,0,2,4,6]), 4 lanes wide; w6 newest.
#define FIR_SHIFT_AND_RP(xv)                                                   \
  w0 = w1; w1 = w2; w2 = w3; w3 = w4; w4 = w5; w5 = w6; w6 = (xv);             \
  const v4f rp = 6.f * (w6 - w0) + 4.f * (w5 - w1) + 2.f * (w4 - w2);

// ---------------------------------------------------------------------------
__global__ void datt_zero_kernel(float* __restrict__ ws) {
  ws[threadIdx.x] = 0.0f;  // 128 threads: [0..63]=sum, [64..127]=sumsq
}

// Pass 1: FIR on the fly, accumulate per-channel sum / sumsq of rp.
__global__ void datt_stats_kernel(const float* __restrict__ x,
                                  float* __restrict__ stats) {
  __shared__ alignas(16) float tile[Tn * POS];   // 64 KB: [t][pos]
  __shared__ float reds[4], redss[4];

  const int tid = threadIdx.x;
  const int c   = blockIdx.y;
  const int p   = (blockIdx.x * BLK + tid) * VEC;  // 0..25084, step 4
  const int b   = p / HWn;
  const int hw  = p - b * HWn;                     // multiple of 4
  const float* __restrict__ xc = x + ((size_t)(b * Cn + c) * Tn) * HWn + hw;

  // 32-deep async prefetch: one b128 per t per thread (512 B/wave contiguous).
#pragma unroll
  for (int t = 0; t < Tn; ++t)
    async_copy_b128(xc + (size_t)t * HWn, &tile[t * POS + tid * VEC]);

  v4f w0 = {0.f,0.f,0.f,0.f}, w1 = w0, w2 = w0, w3 = w0,
      w4 = w0, w5 = w0, w6 = w0;
  v4f s = w0, ss = w0;

#define STAT_GROUP(T0, WN)                                                     \
  wait_asynccnt<WN>();                                                         \
  _Pragma("unroll")                                                            \
  for (int u = 0; u < 8; ++u) {                                                \
    const v4f xv = *(const v4f*)&tile[((T0) + u) * POS + tid * VEC];           \
    FIR_SHIFT_AND_RP(xv)                                                       \
    s += rp; ss += rp * rp;                                                    \
  }
  STAT_GROUP(0, 24)
  STAT_GROUP(8, 16)
  STAT_GROUP(16, 8)
  STAT_GROUP(24, 0)
#undef STAT_GROUP

  float sl  = wave_sum(s.x + s.y + s.z + s.w);
  float ssl = wave_sum(ss.x + ss.y + ss.z + ss.w);
  const int lane = tid & 31, wid = tid >> 5;
  if (lane == 0) { reds[wid] = sl; redss[wid] = ssl; }
  __syncthreads();
  if (tid == 0) {
    float a = 0.f, q = 0.f;
#pragma unroll
    for (int k = 0; k < 4; ++k) { a += reds[k]; q += redss[k]; }
    atomicAdd(&stats[c], a);
    atomicAdd(&stats[Cn + c], q);
  }
}

// Pass 1.5: per-channel scale/bias from accumulated moments.
__global__ void datt_finalize_kernel(const float* __restrict__ gamma,
                                     const float* __restrict__ beta,
                                     float* __restrict__ ws) {
  const int c = threadIdx.x;                 // 64 threads
  const float mean = ws[c] * kInvCount;
  const float var  = fmaxf(ws[Cn + c] * kInvCount - mean * mean, 0.f);
  const float sc   = gamma[c] * rsqrtf(var + kEps);
  ws[2 * Cn + c] = sc;
  ws[3 * Cn + c] = fmaf(-mean, sc, beta[c]); // beta - mean*scale
}

// Pass 2: recompute FIR, BN + ReLU + gate, NT b128 stores.
__global__ void datt_apply_kernel(const float* __restrict__ x,
                                  const float* __restrict__ ws,
                                  const float* __restrict__ rpw,
                                  float* __restrict__ out) {
  __shared__ alignas(16) float tile[Tn * POS];   // 64 KB

  const int tid = threadIdx.x;
  const int c   = blockIdx.y;
  const int p   = (blockIdx.x * BLK + tid) * VEC;
  const int b   = p / HWn;
  const int hw  = p - b * HWn;
  const size_t base = ((size_t)(b * Cn + c) * Tn) * HWn + hw;
  const float* __restrict__ xc = x + base;
  float* __restrict__ oc = out + base;

  const float sc = ws[2 * Cn + c];
  const float bi = ws[3 * Cn + c];
  const float r0 = rpw[0];
  const float r1 = rpw[1];

#pragma unroll
  for (int t = 0; t < Tn; ++t)
    async_copy_b128(xc + (size_t)t * HWn, &tile[t * POS + tid * VEC]);

  v4f w0 = {0.f,0.f,0.f,0.f}, w1 = w0, w2 = w0, w3 = w0,
      w4 = w0, w5 = w0, w6 = w0;
  const v4f vzero = w0;

#define OUT_GROUP(T0, WN)                                                      \
  wait_asynccnt<WN>();                                                         \
  _Pragma("unroll")                                                            \
  for (int u = 0; u < 8; ++u) {                                                \
    const int t = (T0) + u;                                                    \
    const v4f xv = *(const v4f*)&tile[t * POS + tid * VEC];                    \
    FIR_SHIFT_AND_RP(xv)                                                       \
    const v4f bn = sc * rp + bi;                                               \
    const v4f y  = __builtin_elementwise_max(bn, vzero);                       \
    const v4f g  = r1 * (y + 1.f) + r0;     /* rpw0 + rpw1*(y+1) */            \
    __builtin_nontemporal_store(w6 * g, (v4f*)(oc + (size_t)t * HWn));         \
  }
  OUT_GROUP(0, 24)
  OUT_GROUP(8, 16)
  OUT_GROUP(16, 8)
  OUT_GROUP(24, 0)
#undef OUT_GROUP
}

// ---------------------------------------------------------------------------
extern "C" void kernel_launch(void* const* d_in, const int* in_sizes, int n_in,
                              void* d_out, int out_size, void* d_ws, size_t ws_size,
                              hipStream_t stream) {
  const float* x     = (const float*)d_in[0];
  const float* gamma = (const float*)d_in[1];
  const float* beta  = (const float*)d_in[2];
  const float* rpw   = (const float*)d_in[3];
  // d_in[4] is w (==7), baked into the kernels.
  float* ws  = (float*)d_ws;   // [0..63] sum, [64..127] sumsq, [128..191] scale, [192..255] bias
  float* out = (float*)d_out;

  dim3 grid(GRIDX, Cn, 1);
  datt_zero_kernel<<<1, 128, 0, stream>>>(ws);
  datt_stats_kernel<<<grid, BLK, 0, stream>>>(x, ws);
  datt_finalize_kernel<<<1, Cn, 0, stream>>>(gamma, beta, ws);
  datt_apply_kernel<<<grid, BLK, 0, stream>>>(x, ws, rpw, out);
}